// MoEWithAll2All_12240656794283
// MI455X (gfx1250) — compile-verified
//
#include <hip/hip_runtime.h>
#include <hip/hip_bf16.h>
#include <math.h>

#define NE     16      // experts
#define KTOP   2
#define DIMD   1024
#define INTER  704
#define NTOK   4096    // B*S
#define TM     32      // tokens per block tile (2 x 16-row WMMA fragments)
#define HSTRIDE 708    // 704 + 4 pad (704 % 64 == 0 -> bank conflicts otherwise)

typedef __attribute__((ext_vector_type(2))) float v2f;
typedef __attribute__((ext_vector_type(8))) float v8f;

// ---------------------------------------------------------------------------
// Kernel 1: router. One thread per token: 16 dot products over DIM, top-2
// selection on exp-weights (softmax denominator cancels in renormalization),
// build per-expert compact lists via atomicAdd.
// ---------------------------------------------------------------------------
__global__ __launch_bounds__(256) void moe_router_k(
    const float* __restrict__ x, const float* __restrict__ Wg,
    int* __restrict__ cnt, int* __restrict__ tok,
    float* __restrict__ wgt, int* __restrict__ slt)
{
  int n = blockIdx.x * blockDim.x + threadIdx.x;
  if (n >= NTOK) return;

  float acc[NE];
  #pragma unroll
  for (int e = 0; e < NE; ++e) acc[e] = 0.f;

  const float* xr = x + (size_t)n * DIMD;
  for (int d = 0; d < DIMD; ++d) {
    float xv = xr[d];
    #pragma unroll
    for (int e = 0; e < NE; ++e) acc[e] += xv * Wg[e * DIMD + d];
  }

  float m = acc[0];
  #pragma unroll
  for (int e = 1; e < NE; ++e) m = fmaxf(m, acc[e]);
  float p[NE];
  #pragma unroll
  for (int e = 0; e < NE; ++e) p[e] = __expf(acc[e] - m);

  // top-2, earliest index wins ties (matches lax.top_k)
  int i0 = 0; float v0 = p[0];
  #pragma unroll
  for (int e = 1; e < NE; ++e) if (p[e] > v0) { v0 = p[e]; i0 = e; }
  int i1 = (i0 == 0) ? 1 : 0; float v1 = p[i1];
  #pragma unroll
  for (int e = 0; e < NE; ++e) {
    if (e != i0 && p[e] > v1) { v1 = p[e]; i1 = e; }
  }
  float inv = 1.0f / (v0 + v1);          // renormalize over top-2
  float w0 = v0 * inv, w1 = v1 * inv;

  int pos0 = atomicAdd(&cnt[i0], 1);
  tok[i0 * NTOK + pos0] = n;  wgt[i0 * NTOK + pos0] = w0;  slt[i0 * NTOK + pos0] = 0;
  int pos1 = atomicAdd(&cnt[i1], 1);
  tok[i1 * NTOK + pos1] = n;  wgt[i1 * NTOK + pos1] = w1;  slt[i1 * NTOK + pos1] = 1;
}

// fast silu: g * rcp(1 + exp(-g)); v_rcp_f32 approx (~1 ulp), fine here
__device__ __forceinline__ float silu_fast(float g) {
  return g * __builtin_amdgcn_rcpf(1.0f + __expf(-g));
}

// ---------------------------------------------------------------------------
// Kernel 2: per-expert fused SwiGLU GEMM on a 32-token tile using f32 WMMA.
// grid = (NE, NTOK/TM), block = 256 (8 wave32).
// Two 16-row A fragments share each B fragment -> 2x FLOP/byte on weights.
// Phase A: H = silu(X@W1) * (X@W3)  -> LDS      (waves split INTER cols)
// Phase B: O = (H@W2) * gate        -> out_slot (waves split DIM cols)
// ---------------------------------------------------------------------------
__global__ __launch_bounds__(256) void moe_expert_k(
    const float* __restrict__ x,
    const float* __restrict__ W1, const float* __restrict__ W2,
    const float* __restrict__ W3,
    const int* __restrict__ cnt, const int* __restrict__ tok,
    const float* __restrict__ wgt, const int* __restrict__ slt,
    float* __restrict__ out_slot)
{
  const int e     = blockIdx.x;
  const int tile  = blockIdx.y;
  const int count = cnt[e];
  if (tile * TM >= count) return;          // uniform per block -> EXEC stays full

  __shared__ float Hs[TM * HSTRIDE];       // ~90 KB of the 320 KB/WGP budget
  __shared__ int   tokS[TM];
  __shared__ float wS[TM];
  __shared__ int   sltS[TM];

  const int tid = threadIdx.x;
  if (tid < TM) {
    int gi = tile * TM + tid;
    bool v = gi < count;
    tokS[tid] = v ? tok[e * NTOK + gi] : 0;
    wS[tid]   = v ? wgt[e * NTOK + gi] : 0.f;
    sltS[tid] = v ? slt[e * NTOK + gi] : -1;
  }
  __syncthreads();

  const int lane = tid & 31;
  const int wave = tid >> 5;
  const int half = lane >> 4;              // 0: K=0,1  1: K=2,3 (A/B frag layout)
  const int lidx = lane & 15;              // M for A rows, N for B cols
  const int kk   = half * 2;

  const float* W1e  = W1 + (size_t)e * DIMD * INTER;
  const float* W3e  = W3 + (size_t)e * DIMD * INTER;
  const float* W2e  = W2 + (size_t)e * INTER * DIMD;
  const float* xrow0 = x + (size_t)tokS[lidx]      * DIMD;  // A rows 0..15
  const float* xrow1 = x + (size_t)tokS[16 + lidx] * DIMD;  // A rows 16..31

  // ---- Phase A: H = silu(X@W1) * (X@W3) ----
  for (int c = wave; c < INTER / 16; c += 8) {
    const int col0 = c * 16;
    // prefetch next chunk's weight columns into cache ahead of the WMMA stream
    if (c + 8 < INTER / 16) {
      const int coln = (c + 8) * 16 + lidx;
      __builtin_prefetch(W1e + (size_t)kk * INTER + coln, 0, 0);
      __builtin_prefetch(W3e + (size_t)kk * INTER + coln, 0, 0);
    }
    v8f a1_0 = {}; v8f a3_0 = {};          // rows 0..15
    v8f a1_1 = {}; v8f a3_1 = {};          // rows 16..31
    #pragma unroll 2
    for (int k0 = 0; k0 < DIMD; k0 += 4) {
      v2f a0; a0.x = xrow0[k0 + kk]; a0.y = xrow0[k0 + kk + 1];
      v2f a1; a1.x = xrow1[k0 + kk]; a1.y = xrow1[k0 + kk + 1];
      const float* b1p = W1e + (size_t)(k0 + kk) * INTER + col0 + lidx;
      const float* b3p = W3e + (size_t)(k0 + kk) * INTER + col0 + lidx;
      v2f b1; b1.x = b1p[0]; b1.y = b1p[INTER];
      v2f b3; b3.x = b3p[0]; b3.y = b3p[INTER];
      a1_0 = __builtin_amdgcn_wmma_f32_16x16x4_f32(false, a0, false, b1, (short)0, a1_0, false, false);
      a1_1 = __builtin_amdgcn_wmma_f32_16x16x4_f32(false, a1, false, b1, (short)0, a1_1, false, false);
      a3_0 = __builtin_amdgcn_wmma_f32_16x16x4_f32(false, a0, false, b3, (short)0, a3_0, false, false);
      a3_1 = __builtin_amdgcn_wmma_f32_16x16x4_f32(false, a1, false, b3, (short)0, a3_1, false, false);
    }
    #pragma unroll
    for (int i = 0; i < 8; ++i) {
      int r = i + 8 * half;                // C/D layout: VGPR i -> row i (+8 hi half)
      Hs[r * HSTRIDE + col0 + lidx]        = silu_fast(a1_0[i]) * a3_0[i];
      Hs[(16 + r) * HSTRIDE + col0 + lidx] = silu_fast(a1_1[i]) * a3_1[i];
    }
  }
  __syncthreads();

  // ---- Phase B: O = (H @ W2) * gate ----
  for (int c = wave; c < DIMD / 16; c += 8) {
    const int col0 = c * 16;
    if (c + 8 < DIMD / 16) {
      __builtin_prefetch(W2e + (size_t)kk * DIMD + (c + 8) * 16 + lidx, 0, 0);
    }
    v8f acc0 = {}; v8f acc1 = {};
    #pragma unroll 2
    for (int k0 = 0; k0 < INTER; k0 += 4) {
      v2f a0; a0.x = Hs[lidx * HSTRIDE + k0 + kk];
              a0.y = Hs[lidx * HSTRIDE + k0 + kk + 1];
      v2f a1; a1.x = Hs[(16 + lidx) * HSTRIDE + k0 + kk];
              a1.y = Hs[(16 + lidx) * HSTRIDE + k0 + kk + 1];
      const float* bp = W2e + (size_t)(k0 + kk) * DIMD + col0 + lidx;
      v2f b; b.x = bp[0]; b.y = bp[DIMD];
      acc0 = __builtin_amdgcn_wmma_f32_16x16x4_f32(false, a0, false, b, (short)0, acc0, false, false);
      acc1 = __builtin_amdgcn_wmma_f32_16x16x4_f32(false, a1, false, b, (short)0, acc1, false, false);
    }
    #pragma unroll
    for (int i = 0; i < 8; ++i) {
      int r = i + 8 * half;
      int sl0 = sltS[r];
      if (sl0 >= 0) {
        out_slot[(size_t)sl0 * NTOK * DIMD + (size_t)tokS[r] * DIMD + col0 + lidx]
            = wS[r] * acc0[i];
      }
      int sl1 = sltS[16 + r];
      if (sl1 >= 0) {
        out_slot[(size_t)sl1 * NTOK * DIMD + (size_t)tokS[16 + r] * DIMD + col0 + lidx]
            = wS[16 + r] * acc1[i];
      }
    }
  }
}

// ---------------------------------------------------------------------------
// Kernel 3: deterministic combine of the two top-k slot buffers.
// ---------------------------------------------------------------------------
__global__ __launch_bounds__(256) void moe_combine_k(
    const float* __restrict__ s, float* __restrict__ out)
{
  size_t i = (size_t)blockIdx.x * blockDim.x + threadIdx.x;
  if (i < (size_t)NTOK * DIMD) out[i] = s[i] + s[(size_t)NTOK * DIMD + i];
}

// ---------------------------------------------------------------------------
extern "C" void kernel_launch(void* const* d_in, const int* in_sizes, int n_in,
                              void* d_out, int out_size, void* d_ws, size_t ws_size,
                              hipStream_t stream)
{
  const float* x  = (const float*)d_in[0];
  const float* Wg = (const float*)d_in[1];
  const float* W1 = (const float*)d_in[2];
  const float* W2 = (const float*)d_in[3];
  const float* W3 = (const float*)d_in[4];
  float* out = (float*)d_out;

  // workspace layout
  char* ws = (char*)d_ws;
  int*   cnt      = (int*)ws;                               // 16 ints
  int*   tok      = (int*)(ws + 256);                       // NE*NTOK ints
  float* wgt      = (float*)(ws + 256 + NE * NTOK * 4);     // NE*NTOK floats
  int*   slt      = (int*)(ws + 256 + 2 * NE * NTOK * 4);   // NE*NTOK ints
  float* out_slot = (float*)(ws + 256 + 3 * NE * NTOK * 4); // 2*NTOK*DIMD floats

  hipMemsetAsync(cnt, 0, NE * sizeof(int), stream);

  moe_router_k<<<NTOK / 256, 256, 0, stream>>>(x, Wg, cnt, tok, wgt, slt);

  dim3 grid(NE, NTOK / TM);
  moe_expert_k<<<grid, 256, 0, stream>>>(x, W1, W2, W3, cnt, tok, wgt, slt, out_slot);

  moe_combine_k<<<((size_t)NTOK * DIMD + 255) / 256, 256, 0, stream>>>(out_slot, out);
}